// FixedSizeTransformer_31482110280441
// MI455X (gfx1250) — compile-verified
//
#include <hip/hip_runtime.h>

// ---------------------------------------------------------------------------
// Problem constants (reference: N=1e6 rows, D=128 dims, C=1000 classes)
// ---------------------------------------------------------------------------
#define DIMS      128
#define NCLASS    1000
#define CHUNK     32                 // dims per LDS-accumulation chunk
#define NCHUNK    (DIMS / CHUNK)     // 4
#define OUTD      (3 * DIMS)         // 384

typedef float  v2f __attribute__((ext_vector_type(2)));
typedef float  v8f __attribute__((ext_vector_type(8)));
typedef int    v4i __attribute__((vector_size(16)));      // matches builtin param
typedef __attribute__((address_space(1))) v4i* gptr_v4i;
typedef __attribute__((address_space(3))) v4i* lptr_v4i;

// Workspace layout (floats):
//   [0, C*D)                     class sums
//   [C*D, C*D + C)               class counts (float)
//   [C*D + C, C*D + C + D)       global mean
//   [C*D + C + D, ... + C*D)     class means
#define WS_SUMS    0
#define WS_CNT     (NCLASS * DIMS)
#define WS_GMEAN   (NCLASS * DIMS + NCLASS)
#define WS_CMEAN   (NCLASS * DIMS + NCLASS + DIMS)

// ---------------------------------------------------------------------------
// Kernel 0: zero the accumulators (sums + counts) -- must re-zero every call
// ---------------------------------------------------------------------------
__global__ void k_zero(float* __restrict__ ws) {
    int i = blockIdx.x * blockDim.x + threadIdx.x;
    if (i < NCLASS * DIMS + NCLASS) ws[i] = 0.0f;
}

// ---------------------------------------------------------------------------
// Kernel 1: class counts, LDS-aggregated (ds_add_u32) then flushed
// ---------------------------------------------------------------------------
__global__ void k_counts(const int* __restrict__ y, float* __restrict__ ws, int N) {
    __shared__ unsigned int lcnt[NCLASS];
    for (int c = threadIdx.x; c < NCLASS; c += blockDim.x) lcnt[c] = 0u;
    __syncthreads();

    int stride = gridDim.x * blockDim.x;
    for (int i = blockIdx.x * blockDim.x + threadIdx.x; i < N; i += stride)
        atomicAdd(&lcnt[y[i]], 1u);

    __syncthreads();
    float* gcnt = ws + WS_CNT;
    for (int c = threadIdx.x; c < NCLASS; c += blockDim.x) {
        unsigned int v = lcnt[c];
        if (v) atomicAdd(&gcnt[c], (float)v);
    }
}

// ---------------------------------------------------------------------------
// Kernel 2: per-class sums. grid = (NCHUNK, ROWBLK). Each block owns a 32-dim
// chunk and accumulates into a 1000x32 f32 LDS tile (128 KB, conflict-free:
// lanes 0..31 hit 32 consecutive banks), then flushes with global atomics.
// Per row each wave reads one contiguous 128B segment -> coalesced.
// ---------------------------------------------------------------------------
__global__ void k_sums(const float* __restrict__ X, const int* __restrict__ y,
                       float* __restrict__ ws, int N) {
    __shared__ float acc[NCLASS * CHUNK];          // 128000 bytes
    for (int e = threadIdx.x; e < NCLASS * CHUNK; e += blockDim.x) acc[e] = 0.0f;
    __syncthreads();

    const int lane    = threadIdx.x & 31;
    const int wave    = threadIdx.x >> 5;          // 0..7
    const int dimbase = blockIdx.x * CHUNK;
    const int wgl     = blockIdx.y * (blockDim.x >> 5) + wave;
    const int wstride = gridDim.y * (blockDim.x >> 5);

    for (int row = wgl; row < N; row += wstride) {
        int   cls = y[row];
        float v   = X[(size_t)row * DIMS + dimbase + lane];
        atomicAdd(&acc[cls * CHUNK + lane], v);    // ds_add_f32
    }

    __syncthreads();
    float* sums = ws + WS_SUMS;
    for (int e = threadIdx.x; e < NCLASS * CHUNK; e += blockDim.x) {
        float v = acc[e];
        if (v != 0.0f)
            atomicAdd(&sums[(e >> 5) * DIMS + dimbase + (e & 31)], v);
    }
}

// ---------------------------------------------------------------------------
// Kernel 3: global mean = (1/N) * sum over classes of class sums.
// Done as a WMMA reduction: C += ones(16x4) x B(4 classes x 16 dims).
// Row M=0 of C ends up holding the column sums. 8 blocks x 1 wave, 250 steps.
// ---------------------------------------------------------------------------
__global__ void k_gmean(float* __restrict__ ws, float invN) {
    const float* sums  = ws + WS_SUMS;
    float*       gmean = ws + WS_GMEAN;
    const int lane    = threadIdx.x;          // blockDim.x == 32
    const int dimbase = blockIdx.x * 16;      // 8 blocks cover 128 dims

#if __has_builtin(__builtin_amdgcn_wmma_f32_16x16x4_f32)
    // B layout (32-bit, K=4): VGPR0 = K0(l0-15)/K2(l16-31), VGPR1 = K1/K3
    const int n  = lane & 15;
    const int kb = (lane < 16) ? 0 : 2;
    v2f a; a[0] = 1.0f; a[1] = 1.0f;          // A = all-ones 16x4
    v8f c = {};
    for (int k0 = 0; k0 < NCLASS; k0 += 4) {
        v2f b;
        b[0] = sums[(size_t)(k0 + kb + 0) * DIMS + dimbase + n];
        b[1] = sums[(size_t)(k0 + kb + 1) * DIMS + dimbase + n];
        c = __builtin_amdgcn_wmma_f32_16x16x4_f32(
                false, a, false, b, (short)0, c, false, false);
    }
    // C VGPR0: lanes 0..15 hold M=0, N=lane
    if (lane < 16) gmean[dimbase + lane] = c[0] * invN;
#else
    if (lane < 16) {
        float s = 0.0f;
        for (int k = 0; k < NCLASS; ++k) s += sums[(size_t)k * DIMS + dimbase + lane];
        gmean[dimbase + lane] = s * invN;
    }
#endif
}

// ---------------------------------------------------------------------------
// Kernel 4: class means (empty class -> global mean). grid = C, block = D.
// ---------------------------------------------------------------------------
__global__ void k_cmean(float* __restrict__ ws) {
    const int c = blockIdx.x, d = threadIdx.x;
    float cnt = ws[WS_CNT + c];
    float v;
    if (cnt > 0.5f) v = ws[WS_SUMS + (size_t)c * DIMS + d] * (1.0f / cnt);
    else            v = ws[WS_GMEAN + d];
    ws[WS_CMEAN + (size_t)c * DIMS + d] = v;
}

// ---------------------------------------------------------------------------
// Kernel 5: streaming output. One wave per row; each lane moves 3 float4s.
// gmean broadcast into LDS once per block via async-load-to-LDS, then kept
// in registers. ~2 GB of traffic -> the HBM-bound pass.
// ---------------------------------------------------------------------------
__global__ void k_output(const float* __restrict__ X, const int* __restrict__ y,
                         const float* __restrict__ ws, float* __restrict__ out,
                         int N) {
    __shared__ float sgm[DIMS];
    const float* gmean = ws + WS_GMEAN;
    const float* cmean = ws + WS_CMEAN;
    const int lane = threadIdx.x & 31;

    if (threadIdx.x < 32) {   // wave-uniform branch: wave 0 only
#if __has_builtin(__builtin_amdgcn_global_load_async_to_lds_b128)
        __builtin_amdgcn_global_load_async_to_lds_b128(
            (gptr_v4i)(gmean + lane * 4),
            (lptr_v4i)(&sgm[lane * 4]),
            0, 0);
#if __has_builtin(__builtin_amdgcn_s_wait_asynccnt)
        __builtin_amdgcn_s_wait_asynccnt(0);
#else
        asm volatile("s_wait_asynccnt 0x0" ::: "memory");
#endif
#else
        // fallback: plain coalesced copy through VGPRs
        ((float4*)sgm)[lane] = ((const float4*)gmean)[lane];
#endif
    }
    __syncthreads();

    const float4 g4 = ((const float4*)sgm)[lane];

    const int wgl     = blockIdx.x * (blockDim.x >> 5) + (threadIdx.x >> 5);
    const int wstride = gridDim.x * (blockDim.x >> 5);

    for (int row = wgl; row < N; row += wstride) {
        const float* xrow = X + (size_t)row * DIMS;
        // hint the next row this wave will touch (global_prefetch_b8)
        if (row + wstride < N)
            __builtin_prefetch(X + (size_t)(row + wstride) * DIMS + lane * 4, 0, 0);

        const int    cls = y[row];
        const float4 xv  = ((const float4*)xrow)[lane];
        const float4 cv  = ((const float4*)(cmean + (size_t)cls * DIMS))[lane];

        float4* o = (float4*)(out + (size_t)row * OUTD);
        o[lane]      = xv;   // cols   0..127 : X row
        o[32 + lane] = g4;   // cols 128..255 : global mean
        o[64 + lane] = cv;   // cols 256..383 : class mean gather
    }
}

// ---------------------------------------------------------------------------
extern "C" void kernel_launch(void* const* d_in, const int* in_sizes, int n_in,
                              void* d_out, int out_size, void* d_ws, size_t ws_size,
                              hipStream_t stream) {
    const float* X  = (const float*)d_in[0];
    const int*   y  = (const int*)d_in[1];
    float*       out = (float*)d_out;
    float*       ws  = (float*)d_ws;
    const int    N   = in_sizes[1];          // y has N elements

    // 0) zero accumulators
    {
        int total = NCLASS * DIMS + NCLASS;
        k_zero<<<(total + 255) / 256, 256, 0, stream>>>(ws);
    }
    // 1) counts
    k_counts<<<1024, 256, 0, stream>>>(y, ws, N);
    // 2) class sums (4 dim-chunks x 128 row-blocks, 8 waves each)
    {
        dim3 grid(NCHUNK, 128);
        k_sums<<<grid, 256, 0, stream>>>(X, y, ws, N);
    }
    // 3) global mean via WMMA reduction over class sums
    k_gmean<<<DIMS / 16, 32, 0, stream>>>(ws, 1.0f / (float)N);
    // 4) class means with empty-class fallback
    k_cmean<<<NCLASS, DIMS, 0, stream>>>(ws);
    // 5) fused gather + concat streaming write
    k_output<<<4096, 256, 0, stream>>>(X, y, ws, out, N);
}